// QueryAndGroupEquiv_38044820308019
// MI455X (gfx1250) — compile-verified
//
#include <hip/hip_runtime.h>

typedef float v2f __attribute__((ext_vector_type(2)));
typedef float v8f __attribute__((ext_vector_type(8)));

#define B_      4
#define N_      16384
#define NP_     1024
#define C_      64
#define NR_     4
#define NS_     64
#define RADIUS_ 0.2f
#define R2_     (RADIUS_ * RADIUS_)
#define SHADOW_ 1000000.0f

// d_out layout: new_features (B,67,4,Np,64) | grouped_xyz (B,3,Np,64) | bq_idx (B,Np,64)
#define NF_ELEMS ((size_t)B_ * (3 + C_) * NR_ * NP_ * NS_)
#define GX_OFF   NF_ELEMS
#define GX_ELEMS ((size_t)B_ * 3 * NP_ * NS_)
#define BQ_OFF   (GX_OFF + GX_ELEMS)

// ---------------------------------------------------------------------------
// Kernel 1: ball query via V_WMMA_F32_16X16X4_F32.
// d2[m][n] = |q_m|^2 - 2 q_m.p_n + |p_n|^2
//   A (16x4): row m = (qx, qy, qz, 1)
//   B (4x16): col n = (-2px, -2py, -2pz, |p|^2)
//   C (16x16): C[m][n] = |q_m|^2
// One wave handles 16 queries; loops over N in tiles of 16 points.
// D-matrix layout: VGPR v, lanes 0-15 -> row v; lanes 16-31 -> row v+8.
// ---------------------------------------------------------------------------
__launch_bounds__(128)
__global__ void ball_query_wmma(const float* __restrict__ xyz,
                                const float* __restrict__ new_xyz,
                                int* __restrict__ ws_idx) {
    __shared__ float qn2[4][16];
    __shared__ int   lists[4][16][NS_];
    __shared__ int   cnts[4][16];

    const int lane  = threadIdx.x & 31;
    const int wid   = threadIdx.x >> 5;
    const int group = blockIdx.x * 4 + wid;       // 16-query group id
    const int b     = group >> 6;                 // Np/16 = 64 groups per batch
    const int qbase = (group & 63) << 4;

    if (lane < 16) cnts[wid][lane] = 0;

    const int m = lane & 15;
    const float* qp = new_xyz + ((size_t)b * NP_ + qbase + m) * 3;
    const float qx = qp[0], qy = qp[1], qz = qp[2];
    if (lane < 16) qn2[wid][m] = qx * qx + qy * qy + qz * qz;

    // A-matrix (16x4 f32, 2 VGPRs): lanes 0-15 hold K=0,1; lanes 16-31 hold K=2,3
    v2f a;
    a.x = (lane < 16) ? qx : qz;
    a.y = (lane < 16) ? qy : 1.0f;

    __syncthreads();

    // C-matrix: VGPR v = |q_v|^2 (lanes 0-15) / |q_{v+8}|^2 (lanes 16-31)
    v8f cmat;
    for (int v = 0; v < 8; ++v)
        cmat[v] = (lane < 16) ? qn2[wid][v] : qn2[wid][v + 8];

    int nfull = 0;  // meaningful on lane 0 only
    for (int t = 0; t < N_ / 16; ++t) {
        const int pi = (t << 4) + m;
        const float* pp = xyz + ((size_t)b * N_ + pi) * 3;
        const float px = pp[0], py = pp[1], pz = pp[2];

        // B-matrix (4x16 f32, 2 VGPRs): lanes 0-15 hold K=0,1; lanes 16-31 hold K=2,3
        v2f bb;
        bb.x = (lane < 16) ? (-2.0f * px) : (-2.0f * pz);
        bb.y = (lane < 16) ? (-2.0f * py) : (px * px + py * py + pz * pz);

        v8f d = __builtin_amdgcn_wmma_f32_16x16x4_f32(
            /*neg_a=*/false, a, /*neg_b=*/false, bb,
            /*c_mod=*/(short)0, cmat, /*reuse_a=*/false, /*reuse_b=*/false);

        unsigned masks[8];
        for (int v = 0; v < 8; ++v)
            masks[v] = __builtin_amdgcn_ballot_w32(d[v] < R2_);

        if (lane == 0) {
            const int base = t << 4;
            for (int v = 0; v < 8; ++v) {
                for (int h = 0; h < 2; ++h) {
                    const int row = v + (h << 3);
                    unsigned mm = (h == 0) ? (masks[v] & 0xFFFFu) : (masks[v] >> 16);
                    int c = cnts[wid][row];
                    if (c < NS_) {
                        while (mm && c < NS_) {
                            const int bit = __builtin_ctz(mm);
                            mm &= mm - 1;
                            lists[wid][row][c++] = base + bit;
                        }
                        cnts[wid][row] = c;
                        if (c >= NS_) ++nfull;
                    }
                }
            }
        }
        if (__shfl(nfull, 0) == 16) break;  // all 16 rows full: early exit
    }

    __syncthreads();

    // Pad with first hit (or 0 if no hit) and write index table to workspace.
    for (int row = 0; row < 16; ++row) {
        const int cnt   = cnts[wid][row];
        const int first = (cnt > 0) ? lists[wid][row][0] : 0;
        const size_t ob = ((size_t)b * NP_ + qbase + row) * NS_;
        for (int j = lane; j < NS_; j += 32) {
            ws_idx[ob + j] = (j < cnt) ? lists[wid][row][j] : first;
        }
    }
}

// ---------------------------------------------------------------------------
// Kernel 2: grouped_xyz, xyz-feature channels (0..2, broadcast over Nr), bq_idx.
// One 64-thread block per (b, np); thread j = sample slot.
// ---------------------------------------------------------------------------
__launch_bounds__(64)
__global__ void group_xyz(const float* __restrict__ xyz,
                          const float* __restrict__ new_xyz,
                          const int* __restrict__ ws_idx,
                          float* __restrict__ out) {
    const int b  = blockIdx.x >> 10;
    const int np = blockIdx.x & 1023;
    const int j  = threadIdx.x;
    const size_t ib = ((size_t)b * NP_ + np) * NS_;

    const int idx    = ws_idx[ib + j];
    const int first  = ws_idx[ib];
    const bool shadow = (j > 0) && (idx == first);

    const float* pp = xyz + ((size_t)b * N_ + idx) * 3;
    const float* qp = new_xyz + ((size_t)b * NP_ + np) * 3;

    float g[3];
    for (int d = 0; d < 3; ++d) {
        const float p = shadow ? SHADOW_ : pp[d];
        g[d] = p - qp[d];
    }
    // grouped_xyz (B,3,Np,NS)
    for (int d = 0; d < 3; ++d)
        out[GX_OFF + (((size_t)b * 3 + d) * NP_ + np) * NS_ + j] = g[d];
    // xyz features -> new_features channels 0..2, broadcast over Nr
    for (int d = 0; d < 3; ++d) {
        const float f = ((g[d] > 100000.0f) ? 0.0f : g[d]) * (1.0f / RADIUS_);
        for (int r = 0; r < NR_; ++r)
            out[((((size_t)b * (3 + C_) + d) * NR_ + r) * NP_ + np) * NS_ + j] = f;
    }
    // bq_idx (as float, per harness output dtype)
    out[BQ_OFF + ib + j] = (float)idx;
}

// ---------------------------------------------------------------------------
// Kernel 3: grouped features -> new_features channels 3..66.
// One 256-thread block per (b, np): j = tid&63 (sample), r = tid>>6 fixed,
// loop over c. Stores are fully coalesced (64 consecutive floats per j-group).
// ---------------------------------------------------------------------------
__launch_bounds__(256)
__global__ void group_feats(const float* __restrict__ features,
                            const int* __restrict__ ws_idx,
                            float* __restrict__ out) {
    const int b   = blockIdx.x >> 10;
    const int np  = blockIdx.x & 1023;
    const int j   = threadIdx.x & 63;
    const int sub = threadIdx.x >> 6;  // 0..3 == r
    const size_t ib = ((size_t)b * NP_ + np) * NS_;

    const int idx    = ws_idx[ib + j];
    const int first  = ws_idx[ib];
    const bool shadow = (j > 0) && (idx == first);

    // features (B, C, Nr, N): gather column 'idx', stride N between (c,r) rows
    const float* fb = features + (size_t)b * C_ * NR_ * N_ + idx;

    for (int cr = sub; cr < C_ * NR_; cr += 4) {   // cr = c*4 + r, r == sub
        if (cr + 4 < C_ * NR_)
            __builtin_prefetch(&fb[(size_t)(cr + 4) * N_], 0, 1);
        const float v = shadow ? 0.0f : fb[(size_t)cr * N_];
        const int c = cr >> 2, r = cr & 3;
        out[((((size_t)b * (3 + C_) + 3 + c) * NR_ + r) * NP_ + np) * NS_ + j] = v;
    }
}

// ---------------------------------------------------------------------------
extern "C" void kernel_launch(void* const* d_in, const int* in_sizes, int n_in,
                              void* d_out, int out_size, void* d_ws, size_t ws_size,
                              hipStream_t stream) {
    (void)in_sizes; (void)n_in; (void)out_size; (void)ws_size;
    const float* xyz      = (const float*)d_in[0];
    const float* new_xyz  = (const float*)d_in[1];
    const float* features = (const float*)d_in[2];
    float* out   = (float*)d_out;
    int* ws_idx  = (int*)d_ws;   // B*Np*NS ints = 1 MB of scratch

    // Phase 1: WMMA ball query -> index table in workspace
    ball_query_wmma<<<B_ * (NP_ / 16) / 4, 128, 0, stream>>>(xyz, new_xyz, ws_idx);
    // Phase 2: xyz gather + xyz features + bq_idx
    group_xyz<<<B_ * NP_, 64, 0, stream>>>(xyz, new_xyz, ws_idx, out);
    // Phase 3: feature gather (dominant memory traffic, coalesced stores)
    group_feats<<<B_ * NP_, 256, 0, stream>>>(features, ws_idx, out);
}